// SlideWindowConv_69681549410697
// MI455X (gfx1250) — compile-verified
//
#include <hip/hip_runtime.h>
#include <math.h>

// ---------------------------------------------------------------------------
// SlideWindowConv on MI455X (gfx1250, wave32).
// Stage 1: per-(b,s) MLP window weights via V_WMMA_F32_16X16X4_F32.
// Stage 2: bandwidth-bound weighted window sum with LDS tiling.
// ---------------------------------------------------------------------------

typedef __attribute__((ext_vector_type(2))) float v2f;
typedef __attribute__((ext_vector_type(8))) float v8f;

constexpr int NB = 32;
constexpr int NS = 4096;
constexpr int ND = 256;     // channels
constexpr int NWIN = 16;    // window size (15 backward neighbors + self)

// ============================ Stage 1 ======================================
// One wave per (b,s). Rows of the MLP batch = k-offsets 1..15 (+1 dummy) and
// live in the N dimension of the WMMA tiles (transposed formulation):
//   H1^T[32x16] = W1^T[32x3] * F^T[3x16]     (2 WMMAs, K padded 3->4)
//   H2^T[16x16] = W2^T[16x32] * H1^T[32x16]  (8 accumulated WMMAs)
//   w[row]      = sigmoid(W3 . H2^T[:,row] + b3)
// C/D layout: lane l, vgpr v -> (M = v + 8*(l>=16), N = l%16)   [f32 16x16]
// A   layout: lane l, vgpr v -> (M = l%16,  K = v + 2*(l>=16))  [f32 16x4]
// B   layout: lane l, vgpr v -> (K = v + 2*(l>=16), N = l%16)   [f32 4x16]
__global__ __launch_bounds__(256) void slidewin_mlp_weights(
    const float* __restrict__ t,   // [B,S]
    const float* __restrict__ W1,  // [3,32]
    const float* __restrict__ b1,  // [32]
    const float* __restrict__ W2,  // [32,16]
    const float* __restrict__ b2,  // [16]
    const float* __restrict__ W3,  // [16,1]
    const float* __restrict__ b3,  // [1]
    float* __restrict__ wTab)      // [B,S,16]  (slot j = k-1; j==15 dummy=0)
{
    __shared__ float smW[673];         // W1|b1|W2|b2|W3|b3 cache
    __shared__ float stage[8 * 512];   // per-wave 32x16 H1^T staging

    const int tid = threadIdx.x;
    for (int i = tid; i < 673; i += 256) {
        float v;
        if      (i <  96) v = W1[i];
        else if (i < 128) v = b1[i - 96];
        else if (i < 640) v = W2[i - 128];
        else if (i < 656) v = b2[i - 640];
        else if (i < 672) v = W3[i - 656];
        else              v = b3[0];
        smW[i] = v;
    }
    __syncthreads();
    const float* sW1 = smW;          // [3][32]
    const float* sb1 = smW + 96;     // [32]
    const float* sW2 = smW + 128;    // [32][16]
    const float* sb2 = smW + 640;    // [16]
    const float* sW3 = smW + 656;    // [16]
    const float  vb3 = smW[672];

    const int wid  = tid >> 5;
    const int lane = tid & 31;
    const int row  = lane & 15;   // N index (k-1)
    const int hi   = lane >> 4;   // lane half selects K/M sub-block

    const int g = blockIdx.x * 8 + wid;   // flattened (b,s)
    const int b = g / NS;
    const int s = g - b * NS;

    // ---- features: B-matrix (4x16), K = {td, log(td+eps), rel, 0} ----
    const int   k     = row + 1;                 // 1..16 (16 = dummy row)
    const float t_s   = t[g];
    int idx = s - k; idx = idx < 0 ? 0 : idx;    // clamp (no divergence)
    const float t_nb  = t[b * NS + idx];
    const bool  valid = (k <= 15) && (s >= k);
    const float dt  = valid ? fabsf(t_s - t_nb) : 0.0f;
    const float td  = fmaxf(dt, 1e-6f);
    const float ltd = __logf(td + 1e-6f);
    const float cws = fminf((float)(s + 1), 16.0f);
    const float rel = (float)k / cws;

    v2f fb;                         // lanes<16: K0,K1   lanes>=16: K2,K3
    fb.x = hi ? rel : td;
    fb.y = hi ? 0.0f : ltd;

    // ---- layer 1: A = W1^T, bias preloaded into C (constant over N) ----
    v2f a0, a1;
    a0.x = sW1[(2 * hi) * 32 + row];
    a0.y = hi ? 0.0f : sW1[32 + row];
    a1.x = sW1[(2 * hi) * 32 + row + 16];
    a1.y = hi ? 0.0f : sW1[32 + row + 16];

    v8f c0, c1;
    #pragma unroll
    for (int v = 0; v < 8; ++v) {
        c0[v] = sb1[v + 8 * hi];        // hidden 0..15
        c1[v] = sb1[16 + v + 8 * hi];   // hidden 16..31
    }
    v8f h0 = __builtin_amdgcn_wmma_f32_16x16x4_f32(false, a0, false, fb,
                                                   (short)0, c0, false, false);
    v8f h1 = __builtin_amdgcn_wmma_f32_16x16x4_f32(false, a1, false, fb,
                                                   (short)0, c1, false, false);

    // ---- ReLU + stage H1^T[32][16] in wave-local LDS (in-order per wave) ----
    float* st = stage + wid * 512;
    #pragma unroll
    for (int v = 0; v < 8; ++v) {
        const int h = v + 8 * hi;
        st[h * 16 + row]        = fmaxf(h0[v], 0.0f);
        st[(h + 16) * 16 + row] = fmaxf(h1[v], 0.0f);
    }

    // ---- layer 2: 8 accumulated WMMAs over K=32, bias preloaded in C ----
    v8f acc;
    #pragma unroll
    for (int v = 0; v < 8; ++v) acc[v] = sb2[v + 8 * hi];
    #pragma unroll
    for (int c = 0; c < 8; ++c) {
        const int K0 = 4 * c + 2 * hi;
        v2f a, bm;
        a.x  = sW2[K0 * 16 + row];
        a.y  = sW2[(K0 + 1) * 16 + row];
        bm.x = st[K0 * 16 + row];
        bm.y = st[(K0 + 1) * 16 + row];
        acc = __builtin_amdgcn_wmma_f32_16x16x4_f32(false, a, false, bm,
                                                    (short)0, acc, false, false);
    }

    // ---- layer 3: ReLU, 16-dot split across lane halves, sigmoid ----
    float p = 0.0f;
    #pragma unroll
    for (int v = 0; v < 8; ++v) {
        const int h = v + 8 * hi;
        p += fmaxf(acc[v], 0.0f) * sW3[h];
    }
    p += __shfl_xor(p, 16, 32);   // combine hidden 0-7 with 8-15 per row
    float w = 1.0f / (1.0f + __expf(-(p + vb3)));
    if (!valid) w = 0.0f;
    if (hi == 0) wTab[(size_t)g * 16 + row] = w;
}

// ============================ Stage 2 ======================================
// One block per (b, 48-row tile). x tile (48+15 rows x 256 ch) staged in LDS
// as float4; each thread produces float4 outputs for 12 rows. Bandwidth-bound:
// ~1.3x x-read + 1x write  =>  ~14us floor at 23.3 TB/s.
constexpr int TILE_S = 48;
constexpr int HALO   = 15;   // WIN-1
constexpr int ROWS   = TILE_S + HALO;   // 63
constexpr int D4     = ND / 4;          // 64 float4 per row

__global__ __launch_bounds__(256) void slidewin_weighted_sum(
    const float* __restrict__ x,     // [B,S,D]
    const float* __restrict__ wTab,  // [B,S,16]
    float* __restrict__ out)         // [B,S,D]
{
    __shared__ float4 xl[ROWS * D4];   // 64512 B

    const int tid      = threadIdx.x;
    const int numTiles = (NS + TILE_S - 1) / TILE_S;
    const int b        = blockIdx.x / numTiles;
    const int tile     = blockIdx.x - b * numTiles;
    const int s0       = tile * TILE_S;

    const float4* x4 = (const float4*)x;
    for (int i = tid; i < ROWS * D4; i += 256) {
        const int r  = i >> 6;
        const int c  = i & 63;
        const int gs = s0 - HALO + r;
        float4 v = make_float4(0.f, 0.f, 0.f, 0.f);
        if (gs >= 0 && gs < NS)
            v = x4[((size_t)b * NS + gs) * D4 + c];
        xl[i] = v;
    }
    __syncthreads();

    const int d4 = tid & 63;
    const int sg = tid >> 6;
    for (int sl = sg; sl < TILE_S; sl += 4) {
        const int s = s0 + sl;
        if (s >= NS) break;
        const float* wr = wTab + ((size_t)b * NS + s) * 16;
        float wk[15];
        float total = 1.0f;
        #pragma unroll
        for (int j = 0; j < 15; ++j) { wk[j] = wr[j]; total += wk[j]; }

        const int r = sl + HALO;
        float4 a = xl[r * D4 + d4];
        #pragma unroll
        for (int j = 0; j < 15; ++j) {
            const float4 xv = xl[(r - 1 - j) * D4 + d4];
            const float  wv = wk[j];
            a.x = fmaf(wv, xv.x, a.x);
            a.y = fmaf(wv, xv.y, a.y);
            a.z = fmaf(wv, xv.z, a.z);
            a.w = fmaf(wv, xv.w, a.w);
        }
        const float inv = 1.0f / fmaxf(total, 1.0f);
        a.x *= inv; a.y *= inv; a.z *= inv; a.w *= inv;
        ((float4*)out)[((size_t)b * NS + s) * D4 + d4] = a;
    }
}

// ============================ Launch =======================================
extern "C" void kernel_launch(void* const* d_in, const int* in_sizes, int n_in,
                              void* d_out, int out_size, void* d_ws, size_t ws_size,
                              hipStream_t stream) {
    const float* x  = (const float*)d_in[0];
    const float* t  = (const float*)d_in[1];
    const float* W1 = (const float*)d_in[2];
    const float* b1 = (const float*)d_in[3];
    const float* W2 = (const float*)d_in[4];
    const float* b2 = (const float*)d_in[5];
    const float* W3 = (const float*)d_in[6];
    const float* b3 = (const float*)d_in[7];
    float* wTab = (float*)d_ws;              // [B,S,16] f32 = 8 MB
    float* outp = (float*)d_out;

    // Stage 1: one wave per (b,s); 8 waves per block.
    const int nWaves  = NB * NS;             // 131072
    const int blocks1 = nWaves / 8;          // 16384
    slidewin_mlp_weights<<<blocks1, 256, 0, stream>>>(t, W1, b1, W2, b2, W3, b3,
                                                      wTab);

    // Stage 2: (b, tile) blocks.
    const int numTiles = (NS + TILE_S - 1) / TILE_S;   // 86
    slidewin_weighted_sum<<<NB * numTiles, 256, 0, stream>>>(x, wTab, outp);
}